// TensorProduct_13254269075605
// MI455X (gfx1250) — compile-verified
//
#include <hip/hip_runtime.h>

typedef __attribute__((ext_vector_type(2))) float v2f;
typedef __attribute__((ext_vector_type(8))) float v8f;

#define NUM_M  25
#define CCH    128
#define JP     28                  // j-block padded 25 -> 28 (multiple of 4)
#define KP     (NUM_M * JP)        // 700: K = i*28 + j
#define MPAD   32
#define TGS    708                 // Tmat LDS row stride (mult of 4, ==4 mod 64)
#define XS     136                 // x tile LDS row stride (2*136 == 16 mod 64)
#define TG_FLOATS (MPAD * TGS)     // 22656
#define X1_FLOATS (NUM_M * XS)     // 3400
#define X2_FLOATS (JP * XS)        // 3808 (rows 25..27 zeroed)
#define SMEM_BYTES ((TG_FLOATS + X1_FLOATS + X2_FLOATS) * 4)   // 119,456 B

// ---- Kernel 1: zero the dense CG matrix [32 x 700] in workspace ----
__global__ void cg_zero_tmat(float* __restrict__ t) {
  int i = blockIdx.x * 256 + threadIdx.x;
  if (i < MPAD * KP) t[i] = 0.0f;
}

// ---- Kernel 2: scatter COO CG entries: Tmat[seg, M1*28 + M2] += cg ----
__global__ void cg_scatter_tmat(float* __restrict__ t, const float* __restrict__ cg,
                                const int* __restrict__ m1, const int* __restrict__ m2,
                                const int* __restrict__ seg, int nnz) {
  int e = blockIdx.x * 256 + threadIdx.x;
  if (e < nnz) {
    int row = seg[e];
    int col = m1[e] * JP + m2[e];
    atomicAdd(&t[row * KP + col], cg[e]);
  }
}

// ---- Kernel 3: out[b] = Tmat[25 x 700] @ P[700 x 128]
// P[i*28+j, c] = x1[i,c] * x2[j,c]  (j>=25 rows are zero in LDS).
// One block per batch; 8 waves; wave owns 16 columns and two 16x16 f32
// accumulators. Nested i/j loop keeps x1 in a register for 7 WMMA pairs and
// turns every B/A address into a literal ds offset: no div/mod, no cndmask.
__global__ __launch_bounds__(256) void cg_tp_wmma(
    const float* __restrict__ x1, const float* __restrict__ x2,
    const float* __restrict__ tmat, float* __restrict__ out) {
  extern __shared__ float smem[];
  float* tg  = smem;                     // [32][708]
  float* x1s = smem + TG_FLOATS;         // [25][136]
  float* x2s = x1s + X1_FLOATS;          // [28][136]

  const int tid = threadIdx.x;
  const int b   = blockIdx.x;
  const long bbase = (long)b * (NUM_M * CCH);

  // Stage Tmat (zero-padded in global) into LDS, float4 chunks.
  for (int q = tid; q < MPAD * KP / 4; q += 256) {
    int r  = q / (KP / 4);
    int kq = (q % (KP / 4)) * 4;
    float4 v = ((const float4*)tmat)[q];
    *(float4*)&tg[r * TGS + kq] = v;
  }
  // Stage x1/x2 [25 x 128] into padded LDS.
  for (int q = tid; q < NUM_M * CCH / 4; q += 256) {
    int i  = q / (CCH / 4);
    int cq = (q % (CCH / 4)) * 4;
    *(float4*)&x1s[i * XS + cq] = ((const float4*)(x1 + bbase))[q];
    *(float4*)&x2s[i * XS + cq] = ((const float4*)(x2 + bbase))[q];
  }
  // Zero the x2 pad rows (j = 25..27): 3*136 floats.
  for (int q = tid; q < (JP - NUM_M) * XS / 4; q += 256) {
    float4 z = {0.f, 0.f, 0.f, 0.f};
    *(float4*)&x2s[NUM_M * XS + q * 4] = z;
  }
  __syncthreads();

  const int wave = tid >> 5;
  const int lane = tid & 31;
  const int lm   = lane & 15;         // tile-local M row (A) / N col (B,D)
  const int half = lane >> 4;         // K-half selector
  const int c    = wave * 16 + lm;    // channel column owned by this lane

  v8f acc0 = {};                      // m = 0..15
  v8f acc1 = {};                      // m = 16..31 (rows >=25 zero in Tmat)

  const float* tgr0 = &tg[lm * TGS + 2 * half];
  const float* tgr1 = &tg[(lm + 16) * TGS + 2 * half];
  const float* x2p  = &x2s[(2 * half) * XS + c];

  for (int i = 0; i < NUM_M; ++i) {
    const float x1v = x1s[i * XS + c];
    const float* t0 = tgr0 + i * JP;
    const float* t1 = tgr1 + i * JP;
#pragma unroll
    for (int j0 = 0; j0 < JP; j0 += 4) {
      v2f a0 = *(const v2f*)(t0 + j0);                 // ds_load_b64
      v2f a1 = *(const v2f*)(t1 + j0);                 // ds_load_b64
      v2f bv = { x1v * x2p[j0 * XS],                   // literal ds offsets
                 x1v * x2p[j0 * XS + XS] };
      acc0 = __builtin_amdgcn_wmma_f32_16x16x4_f32(false, a0, false, bv,
                                                   (short)0, acc0, false, false);
      acc1 = __builtin_amdgcn_wmma_f32_16x16x4_f32(false, a1, false, bv,
                                                   (short)0, acc1, false, false);
    }
  }

  // D layout: VGPR r -> M = r + 8*half, N = lane%16.
#pragma unroll
  for (int r = 0; r < 8; ++r) {
    int m = r + 8 * half;
    out[bbase + m * CCH + c] = acc0[r];
    int mh = m + 16;
    if (mh < NUM_M) out[bbase + mh * CCH + c] = acc1[r];
  }
}

extern "C" void kernel_launch(void* const* d_in, const int* in_sizes, int n_in,
                              void* d_out, int out_size, void* d_ws, size_t ws_size,
                              hipStream_t stream) {
  const float* x1  = (const float*)d_in[0];
  const float* x2  = (const float*)d_in[1];
  const float* cg  = (const float*)d_in[2];
  const int*   M1  = (const int*)d_in[3];
  const int*   M2  = (const int*)d_in[4];
  const int*   seg = (const int*)d_in[5];
  float* outp = (float*)d_out;

  const int nnz = in_sizes[2];
  const int nB  = in_sizes[0] / (NUM_M * CCH);   // 128
  float* tmat = (float*)d_ws;                     // 32*700*4 = 89.6 KB

  cg_zero_tmat<<<(MPAD * KP + 255) / 256, 256, 0, stream>>>(tmat);
  cg_scatter_tmat<<<(nnz + 255) / 256, 256, 0, stream>>>(tmat, cg, M1, M2, seg, nnz);
  cg_tp_wmma<<<nB, 256, SMEM_BYTES, stream>>>(x1, x2, tmat, outp);
}